// DecodePredictions_80376017977378
// MI455X (gfx1250) — compile-verified
//
#include <hip/hip_runtime.h>
#include <hip/hip_bf16.h>
#include <stdint.h>

// ---------------------------------------------------------------------------
// RetinaNet post-processing for MI455X (gfx1250, wave32).
// predictions: (4, 76725, 84) f32.  Outputs: fb(4,100,4) fs(4,100) fc(4,100,i32) n(4,i32)
// Memory-bound (~103MB in, ~1 flop/byte) -> no WMMA. gfx1250 paths used:
//   * TDM  tensor_load_to_lds (+ s_wait_tensorcnt) for the 128x84 f32 tile DMA in K1
//   * per-lane global_load_async_to_lds_b128 (+ s_wait_asynccnt) for the box gather in K3
// ---------------------------------------------------------------------------

#define NANCH 76725
#define NCLS  80
#define NB    4
#define PRE_K 512
#define PER_CLS 100
#define MAXDET  100
#define TILE  128
#define TILES_PER_IMG 600   // ceil(76725/128)

#define LOGIT_THR  (-2.9444389791664403f)  // logit(0.05): sigmoid(v)>0.05 <=> v>thr

typedef uint32_t u32x4 __attribute__((ext_vector_type(4)));
typedef uint32_t u32x8 __attribute__((ext_vector_type(8)));

__device__ __forceinline__ uint32_t lds_off_u32(const void* p) {
  // low 32 bits of the flat address are the LDS byte offset (CDNA5 aperture rule)
  return (uint32_t)(uint64_t)(uintptr_t)p;
}

// CDNA5 per-lane async copy: 16B global -> LDS, tracked by ASYNCcnt.
__device__ __forceinline__ void async_ld_b128(uint32_t lds_off, uint64_t gaddr) {
  asm volatile("global_load_async_to_lds_b128 %0, %1, off"
               :: "v"(lds_off), "v"(gaddr) : "memory");
}
__device__ __forceinline__ void wait_async0() {
  asm volatile("s_wait_asynccnt 0" ::: "memory");
}

// ---------------------------------------------------------------------------
// TDM: DMA a (rows x 84 f32) 2D tile from global memory into LDS.
// D# per CDNA5 ISA 8.3/8.4:  group0 = {flags, lds_addr, gaddr_lo, gaddr_hi|type},
// group1 packs data_size / tensor dims / tile dims / dim0 stride. tile rows
// beyond tensor_dim1 read as zero (hardware OOB rule) -> ragged last tile free.
// Groups 2/3 zeroed (tile_dim2 == 0 -> higher dims unused).
// ---------------------------------------------------------------------------
__device__ __forceinline__ void tdm_load_tile(uint32_t lds_addr, uint64_t gaddr,
                                              uint32_t rows_remaining) {
  // force SGPR-resident fields (all values are wave-uniform)
  uint32_t ga_lo = __builtin_amdgcn_readfirstlane((uint32_t)gaddr);
  uint32_t ga_hi = __builtin_amdgcn_readfirstlane((uint32_t)(gaddr >> 32));
  uint32_t la    = __builtin_amdgcn_readfirstlane(lds_addr);
  uint32_t td1   = __builtin_amdgcn_readfirstlane(rows_remaining);

  u32x4 g0;
  g0.x = 1u;                                    // count=1 (valid user descriptor)
  g0.y = la;                                    // lds_addr (bytes)
  g0.z = ga_lo;                                 // global_addr[31:0]
  g0.w = (ga_hi & 0x01FFFFFFu) | (2u << 30);    // global_addr[56:32] | type=2
  u32x8 g1;
  g1.s0 = 2u << 16;                             // wg_mask=0 | data_size=2 (4B)
  g1.s1 = 84u << 16;                            // atomic_addr=0 | tensor_dim0[15:0]=84
  g1.s2 = (td1 & 0xFFFFu) << 16;                // tensor_dim0[31:16]=0 | tensor_dim1[15:0]
  g1.s3 = (84u << 16) | ((td1 >> 16) & 0xFFFFu);// tensor_dim1[31:16] | tile_dim0=84
  g1.s4 = (uint32_t)TILE;                       // tile_dim1=128 | tile_dim2=0
  g1.s5 = 84u;                                  // tensor_dim0_stride[31:0]=84
  g1.s6 = 0u;                                   // stride hi | tensor_dim1_stride lo
  g1.s7 = 0u;                                   // tensor_dim1_stride hi
  u32x4 gz = {0u, 0u, 0u, 0u};
  asm volatile("tensor_load_to_lds %0, %1, %2, %3"
               :: "s"(g0), "s"(g1), "s"(gz), "s"(gz)
               : "memory");
}

// Descending bitonic sort of (vals, payload) resident in LDS. len = power of 2.
__device__ __forceinline__ void bitonic_desc(float* vals, int* idx, int len,
                                             int tid, int nthreads) {
  for (int k = 2; k <= len; k <<= 1) {
    for (int j = k >> 1; j > 0; j >>= 1) {
      __syncthreads();
      for (int t = tid; t < len; t += nthreads) {
        int ixj = t ^ j;
        if (ixj > t) {
          bool descend = ((t & k) == 0);
          float a = vals[t], b = vals[ixj];
          if ((a < b) == descend) {
            vals[t] = b; vals[ixj] = a;
            int tmp = idx[t]; idx[t] = idx[ixj]; idx[ixj] = tmp;
          }
        }
      }
    }
  }
  __syncthreads();
}

// ---------------------------------------------------------------------------
// K1: TDM-stage a (128 anchors x 84 floats) tile into LDS, decode boxes,
//     write class logits transposed: scoresT[b][c][n] (coalesced stores).
// ---------------------------------------------------------------------------
__global__ __launch_bounds__(256) void k_decode_transpose(
    const float* __restrict__ pred, float* __restrict__ boxes,
    float* __restrict__ scoresT) {
  __shared__ float tile[TILE * 84];   // 43008 B
  const int tid = threadIdx.x;
  const int b  = blockIdx.x / TILES_PER_IMG;
  const int tt = blockIdx.x % TILES_PER_IMG;
  const int n0 = tt * TILE;

  if (tid < 32) {   // wave 0 issues the tile DMA and waits on its TENSORcnt
    uint64_t ga = (uint64_t)(uintptr_t)(pred + ((size_t)b * NANCH + n0) * 84);
    tdm_load_tile(lds_off_u32(tile), ga, (uint32_t)(NANCH - n0));
    __builtin_amdgcn_s_wait_tensorcnt(0);
  }
  __syncthreads();  // data visible to all 8 waves

  // ---- decode: one thread per anchor ----
  if (tid < TILE) {
    int n = n0 + tid;
    if (n < NANCH) {
      int off, fw, stride;
      if      (n < 57600) { off = 0;     fw = 80; stride = 8;   }
      else if (n < 72000) { off = 57600; fw = 40; stride = 16;  }
      else if (n < 75600) { off = 72000; fw = 20; stride = 32;  }
      else if (n < 76500) { off = 75600; fw = 10; stride = 64;  }
      else                { off = 76500; fw = 5;  stride = 128; }
      int idx  = n - off;
      int a    = idx % 9;
      int cell = idx / 9;
      int gx = cell % fw, gy = cell / fw;
      float fs = (float)stride;
      float cx = (gx + 0.5f) * fs, cy = (gy + 0.5f) * fs;
      float area = (4.0f * fs) * (4.0f * fs);
      const float ratios[3] = {0.5f, 1.0f, 2.0f};
      const float scales[3] = {1.0f, 1.2599210498948732f, 1.5874010519681994f};
      float r  = ratios[a / 3];
      float sc = scales[a % 3];
      float ah = sqrtf(area / r);
      float aw = area / ah;
      aw *= sc; ah *= sc;

      float dx = tile[tid * 84 + 0] * 0.1f;
      float dy = tile[tid * 84 + 1] * 0.1f;
      float dw = tile[tid * 84 + 2] * 0.2f;
      float dh = tile[tid * 84 + 3] * 0.2f;
      float ox = dx * aw + cx;
      float oy = dy * ah + cy;
      float ow = expf(dw) * aw;
      float oh = expf(dh) * ah;

      float* bp = boxes + ((size_t)b * NANCH + n) * 8;
      bp[0] = ox; bp[1] = oy; bp[2] = ow; bp[3] = oh;              // xywh
      bp[4] = ox - ow * 0.5f; bp[5] = oy - oh * 0.5f;              // xyxy
      bp[6] = ox + ow * 0.5f; bp[7] = oy + oh * 0.5f;
    }
  }

  // ---- transpose logits: LDS(strided) -> global(coalesced) ----
  for (int k = 0; k < 40; ++k) {
    int c   = (tid >> 7) + (k << 1);   // 0..79
    int col = tid & 127;
    int n = n0 + col;
    if (n < NANCH)
      scoresT[((size_t)b * NCLS + c) * NANCH + n] = tile[col * 84 + 4 + c];
  }
}

// ---------------------------------------------------------------------------
// K2: per (image,class): top-512 logits via 512-bin histogram + suffix scan,
//     compact, then 512-wide descending bitonic sort in LDS.
// ---------------------------------------------------------------------------
__global__ __launch_bounds__(512) void k_topk512(
    const float* __restrict__ scoresT, float* __restrict__ tkv,
    int* __restrict__ tki) {
  __shared__ int   hist[512];
  __shared__ int   cum[512];
  __shared__ float sv[512];
  __shared__ int   si[512];
  __shared__ int   sT, cntHi, cntT;
  const int tid = threadIdx.x;
  const int g   = blockIdx.x;                 // b*80 + c
  const float* row = scoresT + (size_t)g * NANCH;

  hist[tid] = 0;
  if (tid == 0) { cntHi = 0; cntT = 0; }
  __syncthreads();

  for (int i = tid; i < NANCH; i += 512) {
    float x = row[i];
    int bin = (int)((x + 14.0f) * 16.0f);     // covers [-14, 18) @ 1/16
    bin = bin < 0 ? 0 : (bin > 511 ? 511 : bin);
    atomicAdd(&hist[bin], 1);
  }
  __syncthreads();

  cum[tid] = hist[tid];
  __syncthreads();
  for (int off = 1; off < 512; off <<= 1) {   // inclusive suffix scan
    int v = (tid + off < 512) ? cum[tid + off] : 0;
    __syncthreads();
    cum[tid] += v;
    __syncthreads();
  }
  if (cum[tid] >= PRE_K && (tid == 511 || cum[tid + 1] < PRE_K)) sT = tid;
  __syncthreads();
  const int T = sT;
  const int hiTotal = (T < 511) ? cum[T + 1] : 0;   // < 512

  for (int i = tid; i < NANCH; i += 512) {    // compact (L2-resident re-read)
    float x = row[i];
    int bin = (int)((x + 14.0f) * 16.0f);
    bin = bin < 0 ? 0 : (bin > 511 ? 511 : bin);
    if (bin > T) {
      int p = atomicAdd(&cntHi, 1);
      sv[p] = x; si[p] = i;
    } else if (bin == T) {
      int p = hiTotal + atomicAdd(&cntT, 1);
      if (p < PRE_K) { sv[p] = x; si[p] = i; }
    }
  }
  __syncthreads();

  bitonic_desc(sv, si, PRE_K, tid, 512);

  tkv[(size_t)g * PRE_K + tid] = sv[tid];
  tki[(size_t)g * PRE_K + tid] = si[tid];
}

// ---------------------------------------------------------------------------
// K3: gather top-512 boxes (async global->LDS for the xyxy broadcast array),
//     exact greedy NMS with on-the-fly IoU, in-order compaction to top-100.
// ---------------------------------------------------------------------------
__global__ __launch_bounds__(512) void k_nms(
    const float* __restrict__ boxes, const float* __restrict__ tkv,
    const int* __restrict__ tki, float* __restrict__ cboxes,
    float* __restrict__ cscores) {
  __shared__ float4 sB[512];     // xyxy
  __shared__ int sKeep[512];
  __shared__ int sVld[512];
  __shared__ int rnk[512];
  const int tid = threadIdx.x;
  const int g = blockIdx.x;
  const int b = g / NCLS;

  float v  = tkv[(size_t)g * PRE_K + tid];
  int   id = tki[(size_t)g * PRE_K + tid];
  const float* bp = boxes + ((size_t)b * NANCH + id) * 8;
  // xyxy half of the 32B box record -> straight into LDS (no VGPR staging)
  async_ld_b128(lds_off_u32(&sB[tid]), (uint64_t)(uintptr_t)(bp + 4));
  float w0 = bp[0], w1 = bp[1], w2 = bp[2], w3 = bp[3];          // xywh (regs)

  int vld = (v > LOGIT_THR) ? 1 : 0;
  sKeep[tid] = 1; sVld[tid] = vld;
  int myKeep = 1;
  wait_async0();
  __syncthreads();

  float4 bx = sB[tid];
  float myArea = (bx.z - bx.x) * (bx.w - bx.y);

  for (int i = 0; i < PRE_K - 1; ++i) {
    if (sKeep[i] && sVld[i] && tid > i && myKeep) {
      float4 bi = sB[i];
      float lx = fmaxf(bi.x, bx.x), ly = fmaxf(bi.y, bx.y);
      float rx = fminf(bi.z, bx.z), ry = fminf(bi.w, bx.w);
      float iw = fmaxf(rx - lx, 0.0f), ih = fmaxf(ry - ly, 0.0f);
      float inter = iw * ih;
      float ai = (bi.z - bi.x) * (bi.w - bi.y);
      float iou = inter / fmaxf(ai + myArea - inter, 1e-8f);
      if (iou > 0.5f) { myKeep = 0; sKeep[tid] = 0; }
    }
    __syncthreads();
  }

  int kept = myKeep & vld;
  rnk[tid] = kept;
  __syncthreads();
  for (int off = 1; off < 512; off <<= 1) {   // inclusive prefix scan
    int t = (tid >= off) ? rnk[tid - off] : 0;
    __syncthreads();
    rnk[tid] += t;
    __syncthreads();
  }
  int total = rnk[PRE_K - 1];
  int rank  = rnk[tid] - kept;                // exclusive
  float s = kept ? 1.0f / (1.0f + expf(-v)) : -1.0f;

  if (kept && rank < PER_CLS) {
    cscores[(size_t)g * PER_CLS + rank] = s;
    float* cb = cboxes + ((size_t)g * PER_CLS + rank) * 4;
    cb[0] = w0; cb[1] = w1; cb[2] = w2; cb[3] = w3;
  }
  int tk = total > PER_CLS ? PER_CLS : total;
  if (tid >= tk && tid < PER_CLS) {           // pad remaining slots
    cscores[(size_t)g * PER_CLS + tid] = -1.0f;
    float* cb = cboxes + ((size_t)g * PER_CLS + tid) * 4;
    cb[0] = cb[1] = cb[2] = cb[3] = -1.0f;
  }
}

// ---------------------------------------------------------------------------
// K4: per image, global top-100 of 8000 via histogram threshold + 256-sort.
// ---------------------------------------------------------------------------
__global__ __launch_bounds__(512) void k_final(
    const float* __restrict__ cscores, const float* __restrict__ cboxes,
    float* __restrict__ out_fb, float* __restrict__ out_fs,
    int* __restrict__ out_fc, int* __restrict__ out_n) {
  __shared__ int   hist[512];
  __shared__ int   cum[512];
  __shared__ float cs[256];
  __shared__ int   cp[256];
  __shared__ int   sT, cHi, cT, sCnt;
  const int tid = threadIdx.x;
  const int b = blockIdx.x;
  const float* row = cscores + (size_t)b * NCLS * PER_CLS;   // 8000 contiguous

  hist[tid] = 0;
  if (tid == 0) { cHi = 0; cT = 0; sCnt = 0; }
  if (tid < 256) { cs[tid] = -1e30f; cp[tid] = 0; }
  __syncthreads();

  for (int i = tid; i < NCLS * PER_CLS; i += 512) {
    float s = row[i];
    int bin = (s <= 0.0f) ? 0 : 1 + (int)(s * 510.0f);
    bin = bin > 511 ? 511 : bin;
    atomicAdd(&hist[bin], 1);
  }
  __syncthreads();

  cum[tid] = hist[tid];
  __syncthreads();
  for (int off = 1; off < 512; off <<= 1) {
    int v = (tid + off < 512) ? cum[tid + off] : 0;
    __syncthreads();
    cum[tid] += v;
    __syncthreads();
  }
  if (cum[tid] >= MAXDET && (tid == 511 || cum[tid + 1] < MAXDET)) sT = tid;
  __syncthreads();
  int T = sT;
  int hiTotal = (T < 511) ? cum[T + 1] : 0;

  for (int i = tid; i < NCLS * PER_CLS; i += 512) {
    float s = row[i];
    int bin = (s <= 0.0f) ? 0 : 1 + (int)(s * 510.0f);
    bin = bin > 511 ? 511 : bin;
    if (bin > T) {
      int p = atomicAdd(&cHi, 1);
      if (p < 256) { cs[p] = s; cp[p] = i; }
    } else if (bin == T) {
      int p = hiTotal + atomicAdd(&cT, 1);
      if (p < 256) { cs[p] = s; cp[p] = i; }
    }
  }
  __syncthreads();

  bitonic_desc(cs, cp, 256, tid, 512);

  if (tid < MAXDET) {
    float ts = cs[tid];
    int vm = (ts >= 0.05f) ? 1 : 0;
    int pi = cp[tid];
    int c  = pi / PER_CLS;
    const float* cb = cboxes + ((size_t)b * NCLS * PER_CLS + pi) * 4;
    float* fb = out_fb + ((size_t)b * MAXDET + tid) * 4;
    if (vm) { fb[0] = cb[0]; fb[1] = cb[1]; fb[2] = cb[2]; fb[3] = cb[3]; }
    else    { fb[0] = fb[1] = fb[2] = fb[3] = -1.0f; }
    out_fs[b * MAXDET + tid] = vm ? ts : -1.0f;
    out_fc[b * MAXDET + tid] = vm ? c  : -1;
    if (vm) atomicAdd(&sCnt, 1);
  }
  __syncthreads();
  if (tid == 0) out_n[b] = sCnt;
}

// ---------------------------------------------------------------------------
extern "C" void kernel_launch(void* const* d_in, const int* in_sizes, int n_in,
                              void* d_out, int out_size, void* d_ws, size_t ws_size,
                              hipStream_t stream) {
  const float* pred = (const float*)d_in[1];   // d_in[0] = images (dims only)

  // workspace layout (floats / i32, ~109 MB total)
  float* boxes   = (float*)d_ws;                                   // 4*76725*8
  float* scoresT = boxes   + (size_t)NB * NANCH * 8;               // 4*80*76725
  float* tkv     = scoresT + (size_t)NB * NCLS * NANCH;            // 320*512
  int*   tki     = (int*)(tkv + (size_t)NB * NCLS * PRE_K);        // 320*512
  float* cboxes  = (float*)(tki + (size_t)NB * NCLS * PRE_K);      // 320*100*4
  float* cscores = cboxes + (size_t)NB * NCLS * PER_CLS * 4;       // 320*100

  // output layout (flat concat, 2404 elems): fb | fs | fc(i32) | n(i32)
  float* out_fb = (float*)d_out;
  float* out_fs = out_fb + (size_t)NB * MAXDET * 4;
  int*   out_fc = (int*)(out_fs + (size_t)NB * MAXDET);
  int*   out_n  = out_fc + (size_t)NB * MAXDET;

  k_decode_transpose<<<NB * TILES_PER_IMG, 256, 0, stream>>>(pred, boxes, scoresT);
  k_topk512<<<NB * NCLS, 512, 0, stream>>>(scoresT, tkv, tki);
  k_nms<<<NB * NCLS, 512, 0, stream>>>(boxes, tkv, tki, cboxes, cscores);
  k_final<<<NB, 512, 0, stream>>>(cscores, cboxes, out_fb, out_fs, out_fc, out_n);
}